// ColorRenderer_51307679318771
// MI455X (gfx1250) — compile-verified
//
#include <hip/hip_runtime.h>
#include <math.h>

typedef float v2f __attribute__((ext_vector_type(2)));
typedef float v8f __attribute__((ext_vector_type(8)));

#define IMG 256
#define ORIGF 512.0f
#define PROJ_EPS 1e-9f
#define DEN_EPS 1e-8f
#define MAX_FACE_PAD 1024                 // face capacity
#define AB_HALF (MAX_FACE_PAD * 3)        // float2 entries per LDS half
#define AB_BYTES (AB_HALF * 8)            // 24576 B per half, 48KB total LDS

// ---------------------------------------------------------------------------
// Kernel 1: project vertices to screen space (folds the fv transform in).
// ---------------------------------------------------------------------------
__global__ void project_kernel(const float* __restrict__ verts,
                               const float* __restrict__ K,
                               const float* __restrict__ R,
                               const float* __restrict__ t,
                               const float* __restrict__ dist,
                               float4* __restrict__ outv, int V)
{
    int i = blockIdx.x * blockDim.x + threadIdx.x;
    if (i >= V) return;
    float vx = verts[i*3+0], vy = verts[i*3+1], vz = verts[i*3+2];
    float x = R[0]*vx + R[1]*vy + R[2]*vz + t[0];
    float y = R[3]*vx + R[4]*vy + R[5]*vz + t[1];
    float z = R[6]*vx + R[7]*vy + R[8]*vz + t[2];
    float inv_z = 1.0f / (z + PROJ_EPS);
    float x_ = x * inv_z, y_ = y * inv_z;
    float r2 = x_*x_ + y_*y_;
    float k1 = dist[0], k2 = dist[1], p1 = dist[2], p2 = dist[3], k3 = dist[4];
    float radial = 1.0f + r2*(k1 + r2*(k2 + r2*k3));
    float x__ = x_*radial + 2.0f*p1*x_*y_ + p2*(r2 + 2.0f*x_*x_);
    float y__ = y_*radial + p1*(r2 + 2.0f*y_*y_) + 2.0f*p2*x_*y_;
    float u_pix = K[0]*x__ + K[1]*y__ + K[2];
    float v_pix = K[3]*x__ + K[4]*y__ + K[5];
    float u  = 2.0f*(u_pix - ORIGF*0.5f) / ORIGF;
    float vv = 2.0f*(ORIGF - v_pix - ORIGF*0.5f) / ORIGF;
    float4 o;
    o.x =  u  * (IMG*0.5f) + IMG*0.5f;
    o.y = -vv * (IMG*0.5f) + IMG*0.5f;
    o.z =  z  * (IMG*0.5f) + IMG*0.5f;
    o.w = 0.0f;
    outv[i] = o;
}

// ---------------------------------------------------------------------------
// Kernel 2: per-face affine plane coefficients, split into the exact WMMA-A
// lane layout: AB half holds (a,b) = K0,K1 ; C0 half holds (c,0) = K2,K3.
// Degenerate / padded faces: w0 plane (0,0,-1,0) => never "inside".
// ---------------------------------------------------------------------------
__global__ void plane_kernel(const int* __restrict__ faces,
                             const float4* __restrict__ pv,
                             float2* __restrict__ outAB,
                             float2* __restrict__ outC0,
                             int F, int FT)
{
    int f = blockIdx.x * blockDim.x + threadIdx.x;
    if (f >= FT) return;
    float2 ab0 = make_float2(0.f, 0.f), c0 = make_float2(-1.f, 0.f);
    float2 ab1 = make_float2(0.f, 0.f), c1 = make_float2(-1.f, 0.f);
    float2 abd = make_float2(0.f, 0.f), cd = make_float2(0.f, 0.f);
    if (f < F) {
        int i0 = faces[f*3+0], i1 = faces[f*3+1], i2 = faces[f*3+2];
        float4 a = pv[i0], b = pv[i1], c = pv[i2];
        float denom = (b.y - c.y)*(a.x - c.x) + (c.x - b.x)*(a.y - c.y);
        if (fabsf(denom) > DEN_EPS) {
            float inv = 1.0f / denom;
            float aw0 = (b.y - c.y)*inv, bw0 = (c.x - b.x)*inv;
            float cw0 = -(aw0*c.x + bw0*c.y);
            float aw1 = (c.y - a.y)*inv, bw1 = (a.x - c.x)*inv;
            float cw1 = -(aw1*c.x + bw1*c.y);
            float dz0 = a.z - c.z, dz1 = b.z - c.z;
            ab0 = make_float2(aw0, bw0);               c0 = make_float2(cw0, 0.f);
            ab1 = make_float2(aw1, bw1);               c1 = make_float2(cw1, 0.f);
            abd = make_float2(aw0*dz0 + aw1*dz1,
                              bw0*dz0 + bw1*dz1);
            cd  = make_float2(cw0*dz0 + cw1*dz1 + c.z, 0.f);
        }
    }
    outAB[f*3+0] = ab0;  outC0[f*3+0] = c0;
    outAB[f*3+1] = ab1;  outC0[f*3+1] = c1;
    outAB[f*3+2] = abd;  outC0[f*3+2] = cd;
}

// ---------------------------------------------------------------------------
// Kernel 3: WMMA raster. One wave32 owns 16 pixels; each 16-face tile costs
// 3x ds_load + 3x v_wmma_f32_16x16x4_f32 + VALU min/argmin.
// Planes staged into LDS via GLOBAL_LOAD_ASYNC_TO_LDS_B128 (ASYNCcnt).
// ---------------------------------------------------------------------------
__global__ __launch_bounds__(256)
void raster_kernel(const char* __restrict__ planesBase,  // wsAB .. wsC0, contiguous
                   const float* __restrict__ tex,
                   float* __restrict__ out, int FT)
{
    __shared__ float2 sBuf[2 * AB_HALF];   // [0,AB_BYTES): (a,b)  [AB_BYTES,..): (c,0)

    int tid = threadIdx.x;

    // --- async stage: planes -> LDS, tracked by ASYNCcnt -------------------
    // LDS destination offsets derive from sBuf's real address (low 32 bits of
    // the flat LDS-aperture address = wave-relative LDS offset). The ptrtoint
    // captures sBuf, so the "memory"-clobbering asm is known to write it.
    {
        unsigned ldsBase = (unsigned)(size_t)(void*)sBuf;
        int usedHalf = FT * 3 * 8;                 // bytes per half actually used
        int chunks = usedHalf >> 4;                // b128 chunks per half
        const char* gAB = planesBase;              // (a,b) half
        const char* gC0 = planesBase + AB_BYTES;   // (c,0) half
        for (int i = tid; i < chunks; i += blockDim.x) {
            unsigned off = (unsigned)(i << 4);
            unsigned ldsAB = ldsBase + off;
            unsigned ldsC0 = ldsBase + (unsigned)AB_BYTES + off;
            asm volatile("global_load_async_to_lds_b128 %0, %1, %2"
                         :: "v"(ldsAB), "v"(off), "s"(gAB), "r"((void*)sBuf)
                         : "memory");
            asm volatile("global_load_async_to_lds_b128 %0, %1, %2"
                         :: "v"(ldsC0), "v"(off), "s"(gC0), "r"((void*)sBuf)
                         : "memory");
        }
        asm volatile("s_wait_asynccnt 0" ::: "memory");
    }
    __syncthreads();

    int lane = tid & 31;
    int wave = tid >> 5;
    int n = lane & 15;                         // pixel column within tile
    int p = (blockIdx.x * 8 + wave) * 16 + n;  // global pixel id
    float PX = (float)(p & (IMG - 1)) + 0.5f;
    float PY = (float)(p >> 8) + 0.5f;
    bool hi = lane >= 16;

    // B matrix (4x16): lanes 0-15 hold K=0,1 = (PX,PY); lanes 16-31 K=2,3 = (1,0)
    v2f Bm;
    Bm[0] = hi ? 1.0f : PX;
    Bm[1] = hi ? 0.0f : PY;

    // Lane-half base pointer: lo half reads (a,b), hi half reads (c,0).
    const float2* laneBase = sBuf + (hi ? AB_HALF : 0);
    int faceHalf = hi ? 8 : 0;

    float bestD = INFINITY;
    int bestIdx = 0;

    for (int tile = 0; tile < FT; tile += 16) {
        const float2* tp = laneBase + (tile + n) * 3;
        float2 q0 = tp[0], q1 = tp[1], q2 = tp[2];
        v2f A0, A1, A2;
        A0[0] = q0.x; A0[1] = q0.y;
        A1[0] = q1.x; A1[1] = q1.y;
        A2[0] = q2.x; A2[1] = q2.y;
        v8f zero = {};
        v8f W0 = __builtin_amdgcn_wmma_f32_16x16x4_f32(
            false, A0, false, Bm, (short)0, zero, false, false);
        v8f W1 = __builtin_amdgcn_wmma_f32_16x16x4_f32(
            false, A1, false, Bm, (short)0, zero, false, false);
        v8f DP = __builtin_amdgcn_wmma_f32_16x16x4_f32(
            false, A2, false, Bm, (short)0, zero, false, false);
        int faceBase = tile + faceHalf;
#pragma unroll
        for (int r = 0; r < 8; ++r) {
            float w0 = W0[r], w1 = W1[r];
            float m = fminf(fminf(w0, w1), 1.0f - w0 - w1);   // v_min3 path
            float d = (m >= 0.0f) ? DP[r] : INFINITY;
            int faceId = faceBase + r;
            if (d < bestD) { bestD = d; bestIdx = faceId; }
        }
    }

    // Combine lane halves (faces 0-7 in lanes 0-15, faces 8-15 in lanes 16-31)
    float oD = __shfl_xor(bestD, 16, 32);
    int   oI = __shfl_xor(bestIdx, 16, 32);
    if (oD < bestD || (oD == bestD && oI < bestIdx)) { bestD = oD; bestIdx = oI; }

    if (lane < 16) {
        float cr = 0.f, cg = 0.f, cb = 0.f;
        if (bestD < INFINITY) {
            const float2* ab = sBuf + bestIdx * 3;
            const float2* cc = sBuf + AB_HALF + bestIdx * 3;
            float W0v = ab[0].x*PX + ab[0].y*PY + cc[0].x;
            float W1v = ab[1].x*PX + ab[1].y*PY + cc[1].x;
            float W2v = 1.0f - W0v - W1v;
            const float* tx = tex + bestIdx * 9;   // [vertex][channel]
            cr = W0v*tx[0] + W1v*tx[3] + W2v*tx[6];
            cg = W0v*tx[1] + W1v*tx[4] + W2v*tx[7];
            cb = W0v*tx[2] + W1v*tx[5] + W2v*tx[8];
        }
        out[0*(IMG*IMG) + p] = cr;
        out[1*(IMG*IMG) + p] = cg;
        out[2*(IMG*IMG) + p] = cb;
    }
}

// ---------------------------------------------------------------------------
extern "C" void kernel_launch(void* const* d_in, const int* in_sizes, int n_in,
                              void* d_out, int out_size, void* d_ws, size_t ws_size,
                              hipStream_t stream)
{
    const float* vertices = (const float*)d_in[0];
    const int*   faces    = (const int*)  d_in[1];
    const float* textures = (const float*)d_in[2];
    const float* K        = (const float*)d_in[3];
    const float* R        = (const float*)d_in[4];
    const float* t        = (const float*)d_in[5];
    const float* dist     = (const float*)d_in[6];

    int V = in_sizes[0] / 3;          // 800
    int F = in_sizes[1] / 3;          // 1000
    int FT = (F + 15) & ~15;          // pad to 16-face tiles (1008)
    if (FT > MAX_FACE_PAD) FT = MAX_FACE_PAD;

    // Workspace mirrors the LDS layout: [AB half | C0 half | verts]
    char* ws = (char*)d_ws;
    float2* wsAB    = (float2*)ws;                    // AB_HALF entries
    float2* wsC0    = (float2*)(ws + AB_BYTES);       // AB_HALF entries
    float4* wsVerts = (float4*)(ws + 2 * AB_BYTES);   // V entries

    project_kernel<<<(V + 255) / 256, 256, 0, stream>>>(
        vertices, K, R, t, dist, wsVerts, V);
    plane_kernel<<<(FT + 255) / 256, 256, 0, stream>>>(
        faces, wsVerts, wsAB, wsC0, F, FT);

    int groups = (IMG * IMG) / 16;    // 4096 waves x 16 pixels
    raster_kernel<<<groups / 8, 256, 0, stream>>>(
        ws, textures, (float*)d_out, FT);
}